// DCM_30313879175980
// MI455X (gfx1250) — compile-verified
//
#include <hip/hip_runtime.h>
#include <math.h>

typedef float v2f __attribute__((ext_vector_type(2)));
typedef float v8f __attribute__((ext_vector_type(8)));

#define TAU_F 100.0f
#define TSTR  516           // LDS row stride in floats: 16B aligned, bank-conflict-free
#define NTH   128

__global__ __launch_bounds__(NTH) void dcm_fused_kernel(
    const float* __restrict__ text_feat,   // [64,32,512]
    const float* __restrict__ video_feat,  // [64,12,512]
    const int*   __restrict__ text_mask,   // [64,32]
    float* __restrict__ out)               // [64,64]
{
    __shared__ float t_lds[32 * TSTR];     // normalized text rows
    __shared__ float v_lds[16 * TSTR];     // normalized video rows (12 real + 4 zero pad)
    __shared__ float dot_lds[32 * 16];     // dot[t][v] tile
    __shared__ float tmf[32];
    __shared__ float t2v_lds[32];
    __shared__ float v2t_lds[16];
    __shared__ float vps2_lds[16];

    const int a    = blockIdx.x;
    const int b    = blockIdx.y;
    const int tid  = threadIdx.x;
    const int lane = tid & 31;
    const int wave = tid >> 5;

    if (tid < 32) tmf[tid] = (float)text_mask[a * 32 + tid];

    // ---- stage tiles into LDS (float4, coalesced) ----
    const float4* tsrc = (const float4*)(text_feat + (size_t)a * 32 * 512);
    for (int q = tid; q < 32 * 128; q += NTH) {
        int r = q >> 7, c = q & 127;
        *(float4*)&t_lds[r * TSTR + c * 4] = tsrc[q];
    }
    const float4* vsrc = (const float4*)(video_feat + (size_t)b * 12 * 512);
    for (int q = tid; q < 12 * 128; q += NTH) {
        int r = q >> 7, c = q & 127;
        *(float4*)&v_lds[r * TSTR + c * 4] = vsrc[q];
    }
    for (int q = tid; q < 4 * 128; q += NTH) {           // zero-pad rows 12..15
        int r = 12 + (q >> 7), c = q & 127;
        *(float4*)&v_lds[r * TSTR + c * 4] = float4{0.f, 0.f, 0.f, 0.f};
    }
    __syncthreads();

    // ---- L2-normalize rows (one row per thread) ----
    if (tid < 44) {
        float* row = (tid < 32) ? &t_lds[tid * TSTR] : &v_lds[(tid - 32) * TSTR];
        float s = 0.f;
        const float4* r4 = (const float4*)row;
        for (int c = 0; c < 128; ++c) {
            float4 x = r4[c];
            s += x.x * x.x + x.y * x.y + x.z * x.z + x.w * x.w;
        }
        float inv = 1.0f / fmaxf(sqrtf(s), 1e-6f);
        float4* w4 = (float4*)row;
        for (int c = 0; c < 128; ++c) {
            float4 x = w4[c];
            x.x *= inv; x.y *= inv; x.z *= inv; x.w *= inv;
            w4[c] = x;
        }
    }
    __syncthreads();

    // ---- WMMA: dot[t][v], wave w owns rows 16w..16w+15, 128 x K=4 steps ----
    if (wave < 2) {
        const int half = lane >> 4;            // 0: K+0..1, 1: K+2..3
        const int l15  = lane & 15;
        const float* arow = &t_lds[(wave * 16 + l15) * TSTR + 2 * half];
        const float* brow = &v_lds[l15 * TSTR + 2 * half];
        v8f acc = {};
#pragma unroll 4
        for (int k = 0; k < 512; k += 4) {
            v2f av = *(const v2f*)(arow + k);
            v2f bv = *(const v2f*)(brow + k);
            acc = __builtin_amdgcn_wmma_f32_16x16x4_f32(
                false, av, false, bv, (short)0, acc, false, false);
        }
#pragma unroll
        for (int j = 0; j < 8; ++j) {
            int m = wave * 16 + j + (half << 3);   // lanes 16-31 hold M = j+8
            dot_lds[m * 16 + l15] = acc[j];
        }
    }
    __syncthreads();

    // ---- epilogue ----
    float drow[12];
    float tps2 = 0.f;

    if (wave == 0) {                            // per-row: vps1 softmax -> t2v[t]
        float tmv = tmf[lane];
#pragma unroll
        for (int v = 0; v < 12; ++v) drow[v] = dot_lds[lane * 16 + v];
        float m = -INFINITY;
#pragma unroll
        for (int v = 0; v < 12; ++v) m = fmaxf(m, tmv * drow[v] * TAU_F);
        float s = 0.f, acc = 0.f;
#pragma unroll
        for (int v = 0; v < 12; ++v) {
            float e = expf(tmv * drow[v] * TAU_F - m);
            s += e; acc += e * drow[v];
        }
        t2v_lds[lane] = acc / s;
    } else if (wave == 1 && lane < 12) {        // per-col: tps1 masked softmax -> v2t[v]
        int v = lane;
        float mx = -INFINITY;
        for (int t = 0; t < 32; ++t) {
            float mk = tmf[t] * dot_lds[t * 16 + v];
            float z = (mk == 0.f) ? -INFINITY : mk * TAU_F;
            mx = fmaxf(mx, z);
        }
        float s = 0.f, acc = 0.f;
        for (int t = 0; t < 32; ++t) {
            float dd = dot_lds[t * 16 + v];
            float mk = tmf[t] * dd;
            float z = (mk == 0.f) ? -INFINITY : mk * TAU_F;
            float e = expf(z - mx);
            s += e; acc += e * dd;
        }
        v2t_lds[v] = acc / s;
    }
    __syncthreads();

    if (wave == 0) {                            // tps2 = softmax_t(where(t2v==0,-inf)*tau)
        float x = t2v_lds[lane];
        float z = (x == 0.f) ? -INFINITY : x * TAU_F;
        float m = z;
#pragma unroll
        for (int o = 16; o > 0; o >>= 1) m = fmaxf(m, __shfl_xor(m, o, 32));
        float e = expf(z - m);
        float s = e;
#pragma unroll
        for (int o = 16; o > 0; o >>= 1) s += __shfl_xor(s, o, 32);
        tps2 = e / s;
    }
    if (wave == 1 && lane < 12) {               // vps2 = softmax_v(v2t*tau)
        float m = -INFINITY;
#pragma unroll
        for (int v = 0; v < 12; ++v) m = fmaxf(m, v2t_lds[v] * TAU_F);
        float s = 0.f;
#pragma unroll
        for (int v = 0; v < 12; ++v) s += expf(v2t_lds[v] * TAU_F - m);
        vps2_lds[lane] = expf(v2t_lds[lane] * TAU_F - m) / s;
    }
    __syncthreads();

    if (wave == 0) {                            // out = sum_{t,v} tps2[t]*vps2[v]*dot[t,v]
        float inner = 0.f;
#pragma unroll
        for (int v = 0; v < 12; ++v) inner += vps2_lds[v] * drow[v];
        float part = tps2 * inner;
#pragma unroll
        for (int o = 16; o > 0; o >>= 1) part += __shfl_xor(part, o, 32);
        if (lane == 0) out[a * 64 + b] = part;
    }
}

extern "C" void kernel_launch(void* const* d_in, const int* in_sizes, int n_in,
                              void* d_out, int out_size, void* d_ws, size_t ws_size,
                              hipStream_t stream) {
    (void)in_sizes; (void)n_in; (void)d_ws; (void)ws_size; (void)out_size;
    const float* text_feat  = (const float*)d_in[0];
    const float* video_feat = (const float*)d_in[1];
    const int*   text_mask  = (const int*)d_in[2];
    float* out = (float*)d_out;
    dim3 grid(64, 64, 1);
    dcm_fused_kernel<<<grid, NTH, 0, stream>>>(text_feat, video_feat, text_mask, out);
}